// Voxelization_45354854646368
// MI455X (gfx1250) — compile-verified
//
#include <hip/hip_runtime.h>

#define RES  32
#define R3   (RES * RES * RES)   // 32768
#define NB   32                  // batches
#define NC   64                  // channels
#define NP   8192                // points per batch

typedef __attribute__((ext_vector_type(2))) float v2f;
typedef __attribute__((ext_vector_type(8))) float v8f;

// ---------------------------------------------------------------------------
// Zero the per-voxel counts (4 MiB in d_ws). Must run every call (atomics
// accumulate into it). Voxel output no longer needs zeroing: the accumulate
// kernel writes every element from LDS.
// ---------------------------------------------------------------------------
__global__ __launch_bounds__(256) void vz_zero_kernel(float4* __restrict__ cnt,
                                                      int ncnt4) {
    const float4 z = make_float4(0.f, 0.f, 0.f, 0.f);
    int i = blockIdx.x * blockDim.x + threadIdx.x;
    int stride = gridDim.x * blockDim.x;
    for (int j = i; j < ncnt4; j += stride) cnt[j] = z;
}

// ---------------------------------------------------------------------------
// Per-batch stats: mean(x,y,z) via WMMA ones-matmul reduction, then max L2
// norm of centered coords. One block (256 thr = 8 waves) per batch.
// stats[b*4 + {0,1,2}] = mean, stats[b*4 + 3] = 1/(2*max_norm)
// ---------------------------------------------------------------------------
__global__ __launch_bounds__(256) void vz_stats_kernel(const float* __restrict__ coords,
                                                       float* __restrict__ stats) {
    const int b    = blockIdx.x;
    const int tid  = threadIdx.x;
    const int wave = tid >> 5;
    const int lane = tid & 31;

    const float* xs = coords + (size_t)b * 3 * NP;
    const float* ys = xs + NP;
    const float* zs = xs + 2 * NP;

    __shared__ float redx[8], redy[8], redz[8];
    __shared__ float meanS[3];
    __shared__ float maxred[256];

    // ---- Phase A: sums via V_WMMA_F32_16X16X4_F32 with B = ones ----
    // D += A * Ones  =>  every column of D holds the row-sums of all A tiles.
    v8f accx = {}; v8f accy = {}; v8f accz = {};
    const v2f ones = {1.0f, 1.0f};
    const int base = wave * (NP / 8);          // 1024 points per wave
    #pragma unroll 4
    for (int it = 0; it < 16; ++it) {          // 16 * 64 = 1024 points
        const int n0 = base + it * 64;
        v2f a;
        a.x = xs[n0 + lane]; a.y = xs[n0 + 32 + lane];
        accx = __builtin_amdgcn_wmma_f32_16x16x4_f32(false, a, false, ones,
                                                     (short)0, accx, false, false);
        a.x = ys[n0 + lane]; a.y = ys[n0 + 32 + lane];
        accy = __builtin_amdgcn_wmma_f32_16x16x4_f32(false, a, false, ones,
                                                     (short)0, accy, false, false);
        a.x = zs[n0 + lane]; a.y = zs[n0 + 32 + lane];
        accz = __builtin_amdgcn_wmma_f32_16x16x4_f32(false, a, false, ones,
                                                     (short)0, accz, false, false);
    }
    // Column j of D = rows 0-7 (lane j) + rows 8-15 (lane j+16).
    float lx = 0.f, ly = 0.f, lz = 0.f;
    #pragma unroll
    for (int v = 0; v < 8; ++v) { lx += accx[v]; ly += accy[v]; lz += accz[v]; }
    lx += __shfl_xor(lx, 16, 32);
    ly += __shfl_xor(ly, 16, 32);
    lz += __shfl_xor(lz, 16, 32);
    if (lane == 0) { redx[wave] = lx; redy[wave] = ly; redz[wave] = lz; }
    __syncthreads();
    if (tid == 0) {
        float sx = 0.f, sy = 0.f, sz = 0.f;
        for (int w = 0; w < 8; ++w) { sx += redx[w]; sy += redy[w]; sz += redz[w]; }
        meanS[0] = sx * (1.0f / NP);
        meanS[1] = sy * (1.0f / NP);
        meanS[2] = sz * (1.0f / NP);
    }
    __syncthreads();

    // ---- Phase B: max ||p - mean||^2 ----
    const float mx = meanS[0], my = meanS[1], mz = meanS[2];
    float m2 = 0.0f;
    for (int n = tid; n < NP; n += 256) {
        float dx = xs[n] - mx, dy = ys[n] - my, dz = zs[n] - mz;
        m2 = fmaxf(m2, dx * dx + dy * dy + dz * dz);
    }
    maxred[tid] = m2;
    __syncthreads();
    for (int s = 128; s > 0; s >>= 1) {
        if (tid < s) maxred[tid] = fmaxf(maxred[tid], maxred[tid + s]);
        __syncthreads();
    }
    if (tid == 0) {
        float inv = 1.0f / (2.0f * sqrtf(maxred[0]));   // EPS == 0
        stats[b * 4 + 0] = mx;
        stats[b * 4 + 1] = my;
        stats[b * 4 + 2] = mz;
        stats[b * 4 + 3] = inv;
    }
}

// ---------------------------------------------------------------------------
// Per-point pass: write norm_coords output, store the flat voxel index
// (0..32767, u32) into d_ws, and bump the per-voxel count (u32 hw atomic,
// L2-resident 4 MiB).
// ---------------------------------------------------------------------------
__global__ __launch_bounds__(256) void vz_points_kernel(const float* __restrict__ coords,
                                                        const float* __restrict__ stats,
                                                        float* __restrict__ norm_out,
                                                        unsigned* __restrict__ idxbuf,
                                                        unsigned* __restrict__ counts) {
    const int p = blockIdx.x * 256 + threadIdx.x;   // 0 .. B*N-1
    const int b = p >> 13;                          // / NP
    const int n = p & (NP - 1);

    const float mx  = stats[b * 4 + 0];
    const float my  = stats[b * 4 + 1];
    const float mz  = stats[b * 4 + 2];
    const float inv = stats[b * 4 + 3];

    const float* cb = coords + (size_t)b * 3 * NP;
    const float x = cb[n], y = cb[NP + n], z = cb[2 * NP + n];

    const float sx = fminf(fmaxf(((x - mx) * inv + 0.5f) * (float)RES, 0.0f), (float)(RES - 1));
    const float sy = fminf(fmaxf(((y - my) * inv + 0.5f) * (float)RES, 0.0f), (float)(RES - 1));
    const float sz = fminf(fmaxf(((z - mz) * inv + 0.5f) * (float)RES, 0.0f), (float)(RES - 1));

    float* nb = norm_out + (size_t)b * 3 * NP;
    nb[n]          = sx;
    nb[NP + n]     = sy;
    nb[2 * NP + n] = sz;

    const int vx = (int)rintf(sx);   // v_rndne: round-half-even == jnp.round
    const int vy = (int)rintf(sy);
    const int vz = (int)rintf(sz);
    const unsigned flat = (unsigned)((vx * RES + vy) * RES + vz);

    idxbuf[p] = flat;
    atomicAdd(&counts[b * R3 + flat], 1u);
}

// ---------------------------------------------------------------------------
// One workgroup per (batch, channel): accumulate that channel's 32^3 voxel
// grid in a 128 KiB LDS array with ds_add_f32 atomics (2 workgroups fit per
// 320 KiB WGP), then stream it out with coalesced b128 stores, scaling by
// 1/max(count,1). No global fp atomics, no zero/finalize passes over the
// 268 MB output.
// ---------------------------------------------------------------------------
__global__ __launch_bounds__(256) void vz_accum_kernel(const float* __restrict__ features,
                                                       const unsigned* __restrict__ idxbuf,
                                                       const unsigned* __restrict__ counts,
                                                       float* __restrict__ vox_out) {
    __shared__ float acc[R3];                       // 128 KiB

    const int tid = threadIdx.x;
    const int b   = blockIdx.x >> 6;                // / NC
    const int c   = blockIdx.x & (NC - 1);

    // zero LDS accumulator (b128 stores)
    float4* acc4 = (float4*)acc;
    const float4 z = make_float4(0.f, 0.f, 0.f, 0.f);
    #pragma unroll
    for (int i = tid; i < R3 / 4; i += 256) acc4[i] = z;
    __syncthreads();

    // scatter 8192 points of this (b, c) row into LDS
    const float*    fb = features + ((size_t)b * NC + c) * NP;
    const unsigned* ib = idxbuf + (size_t)b * NP;
    #pragma unroll 4
    for (int n = tid; n < NP; n += 256)
        atomicAdd(&acc[ib[n]], fb[n]);
    __syncthreads();

    // write out, scaled by 1/max(cnt,1); counts stay hot in L2 (reused 64x)
    const uint4* cnt4 = (const uint4*)(counts + (size_t)b * R3);
    float4* out4 = (float4*)(vox_out + ((size_t)b * NC + c) * R3);
    for (int i = tid; i < R3 / 4; i += 256) {
        uint4  cn = cnt4[i];
        float4 a  = acc4[i];
        a.x *= 1.0f / fmaxf((float)cn.x, 1.0f);
        a.y *= 1.0f / fmaxf((float)cn.y, 1.0f);
        a.z *= 1.0f / fmaxf((float)cn.z, 1.0f);
        a.w *= 1.0f / fmaxf((float)cn.w, 1.0f);
        out4[i] = a;
    }
}

// ---------------------------------------------------------------------------
extern "C" void kernel_launch(void* const* d_in, const int* in_sizes, int n_in,
                              void* d_out, int out_size, void* d_ws, size_t ws_size,
                              hipStream_t stream) {
    const float* features = (const float*)d_in[0];   // [B, C, N]
    const float* coords   = (const float*)d_in[1];   // [B, 3, N]

    float* vox_out  = (float*)d_out;                         // [B, C, R, R, R]
    float* norm_out = vox_out + (size_t)NB * NC * R3;        // [B, 3, N]

    float*    stats  = (float*)d_ws;                                   // 512 B
    unsigned* counts = (unsigned*)((char*)d_ws + 1024);                // B*R3 u32 = 4 MiB
    unsigned* idxbuf = (unsigned*)((char*)d_ws + 1024 + (size_t)NB * R3 * 4); // B*N u32 = 1 MiB

    vz_zero_kernel<<<512, 256, 0, stream>>>((float4*)counts, (NB * R3) / 4);
    vz_stats_kernel<<<NB, 256, 0, stream>>>(coords, stats);
    vz_points_kernel<<<(NB * NP) / 256, 256, 0, stream>>>(coords, stats, norm_out,
                                                          idxbuf, counts);
    vz_accum_kernel<<<NB * NC, 256, 0, stream>>>(features, idxbuf, counts, vox_out);
}